// GAT_4879082848675
// MI455X (gfx1250) — compile-verified
//
#include <hip/hip_runtime.h>
#include <cstddef>

// ---------------------------------------------------------------------------
// GAT forward for MI455X (gfx1250): fp32 WMMA GEMM with async-to-LDS staging
// + 3-pass edge softmax (atomics land in the 192MB L2).
// ---------------------------------------------------------------------------

typedef __attribute__((ext_vector_type(2))) float v2f;
typedef __attribute__((ext_vector_type(8))) float v8f;

#define NEG_SLOPE 0.2f

// --------------------------- helpers ---------------------------------------

// Monotonic float atomic max via sign-magnitude int ordering.
// Init value must be -INF (0xFF800000). Positive values use signed-int max,
// negative values use unsigned-int min; both monotone in float order and
// cross-path safe (uint(neg) > uint(pos), int(neg) < int(pos)).
__device__ __forceinline__ void atomicMaxF(float* addr, float val) {
    if (val >= 0.0f) {
        atomicMax((int*)addr, __float_as_int(val));
    } else {
        atomicMin((unsigned int*)addr, __float_as_uint(val));
    }
}

// One wave-wide async copy: each lane contributes a 16B (global -> LDS)
// transfer; tracked on ASYNCcnt, no VGPR bounce, no per-load stall.
__device__ __forceinline__ void async_copy_b128(unsigned lds_addr,
                                                const float* gptr) {
    unsigned long long ga = (unsigned long long)gptr;
    asm volatile("global_load_async_to_lds_b128 %0, %1, off"
                 :: "v"(lds_addr), "v"(ga) : "memory");
}

__device__ __forceinline__ void wait_async0() {
    asm volatile("s_wait_asynccnt 0x0" ::: "memory");
}

// --------------------------- GEMM: feat = h @ W ----------------------------
// h: [n,128], W: [128,128] row-major (in-dim x out-dim), feat: [n,128]
// Block = 256 threads (8 wave32); block tile = 128 rows x 128 cols.
// Wave w owns rows [16w,16w+16) x 128 cols -> 8 v8f accumulators.
// K chunked by 32 through LDS; V_WMMA_F32_16X16X4_F32 does K=4 per op
// (exact fp32: GEMM is tiny vs the edge phase, no reason to drop precision).

#define TK 32

__global__ __launch_bounds__(256) void gemm128_wmma(
    const float* __restrict__ A, const float* __restrict__ W,
    float* __restrict__ C, int n) {
    // sA row stride 36 dwords: rows 16B-aligned for async b128; fragment reads
    // at banks (36*l16 + k) -> 16 distinct bank pairs, conflict-free; aligned
    // b64 A-fragment. sB row stride 144: 16B-aligned rows, 2-row K shift lands
    // the two lane-halves 32 banks apart.
    __shared__ float sA[128][36];
    __shared__ float sB[TK][144];

    const int tid  = threadIdx.x;
    const int lane = tid & 31;
    const int wv   = tid >> 5;          // 0..7
    const int half = lane >> 4;         // 0|1 (lane half selects K pair)
    const int l16  = lane & 15;
    const int row0 = blockIdx.x * 128;
    const bool full = (row0 + 128 <= n);   // block-uniform

    v8f acc[8] = {};                    // 8 tiles x 8 VGPRs f32

    for (int kc = 0; kc < 128; kc += TK) {
        if (full) {
            // Fire-and-forget: 8 async instructions per wave per chunk, each
            // moving 512B straight into LDS; one ASYNCcnt wait at the end.
            #pragma unroll
            for (int i = 0; i < 4; ++i) {
                int slot = tid + 256 * i;
                // A chunk: 128 rows x 32 cols (8 float4 per row)
                int ra = slot >> 3, ca = (slot & 7) * 4;
                async_copy_b128((unsigned)(size_t)&sA[ra][ca],
                                A + (size_t)(row0 + ra) * 128 + kc + ca);
                // W chunk: 32 rows x 128 cols (32 float4 per row)
                int rb = slot >> 5, cb = (slot & 31) * 4;
                async_copy_b128((unsigned)(size_t)&sB[rb][cb],
                                W + (size_t)(kc + rb) * 128 + cb);
            }
            wait_async0();
        } else {
            // Boundary tile: bounce through VGPRs with zero fill for OOB rows.
            float4 va[4], vb[4];
            #pragma unroll
            for (int i = 0; i < 4; ++i) {
                int slot = tid + 256 * i;
                int ra = slot >> 3, ca = (slot & 7) * 4;
                int gr = row0 + ra;
                va[i] = make_float4(0.f, 0.f, 0.f, 0.f);
                if (gr < n) va[i] = *(const float4*)(A + (size_t)gr * 128 + kc + ca);
                int rb = slot >> 5, cb = (slot & 31) * 4;
                vb[i] = *(const float4*)(W + (size_t)(kc + rb) * 128 + cb);
            }
            #pragma unroll
            for (int i = 0; i < 4; ++i) {
                int slot = tid + 256 * i;
                int ra = slot >> 3, ca = (slot & 7) * 4;
                *(float4*)&sA[ra][ca] = va[i];
                int rb = slot >> 5, cb = (slot & 31) * 4;
                *(float4*)&sB[rb][cb] = vb[i];
            }
        }
        __syncthreads();

        const int m0 = wv * 16;
        #pragma unroll
        for (int k = 0; k < TK; k += 4) {
            // A frag 16x4 f32: VGPR0 = K=2*half, VGPR1 = K=2*half+1 (ISA 7.12.2)
            v2f afrag = *(const v2f*)&sA[m0 + l16][k + 2 * half];
            #pragma unroll
            for (int t = 0; t < 8; ++t) {
                // B frag 4x16 f32, symmetric layout: lane = N, halves = K pairs
                v2f bfrag;
                bfrag.x = sB[k + 2 * half + 0][t * 16 + l16];
                bfrag.y = sB[k + 2 * half + 1][t * 16 + l16];
                acc[t] = __builtin_amdgcn_wmma_f32_16x16x4_f32(
                    false, afrag, false, bfrag, (short)0, acc[t], false, false);
            }
        }
        __syncthreads();
    }

    // C/D layout: VGPR j -> (M = j + 8*half, N = l16) within each 16x16 tile
    const int m0 = wv * 16;
    #pragma unroll
    for (int t = 0; t < 8; ++t) {
        #pragma unroll
        for (int j = 0; j < 8; ++j) {
            int r = row0 + m0 + j + 8 * half;
            if (r < n) C[(size_t)r * 128 + t * 16 + l16] = acc[t][j];
        }
    }
}

// ------------------- el/er = einsum('nhd,hd->nh') ---------------------------

__global__ void scores_kernel(const float* __restrict__ feat,
                              const float* __restrict__ al,
                              const float* __restrict__ ar,
                              float* __restrict__ el, float* __restrict__ er,
                              int n, int H, int D) {
    int tid = blockIdx.x * blockDim.x + threadIdx.x;
    if (tid >= n * H) return;
    int node = tid / H, h = tid % H;
    const float* f = feat + (size_t)node * 128 + h * D;
    const float* a = al + h * D;
    const float* b = ar + h * D;
    float sl = 0.f, sr = 0.f;
    #pragma unroll 8
    for (int d = 0; d < D; ++d) { float v = f[d]; sl += v * a[d]; sr += v * b[d]; }
    el[tid] = sl;
    er[tid] = sr;
}

// --------- init: h_next = h + bias (residual), m = -inf, s = 0 --------------

__global__ void init_kernel(const float* __restrict__ h,
                            const float* __restrict__ bias,
                            float* __restrict__ hn,
                            float* __restrict__ m, float* __restrict__ s,
                            int n, int H) {
    size_t tid = (size_t)blockIdx.x * blockDim.x + threadIdx.x;
    size_t tot = (size_t)n * 128;
    if (tid < tot) hn[tid] = h[tid] + bias[tid & 127];
    if (tid < (size_t)n * H) { m[tid] = -INFINITY; s[tid] = 0.f; }
}

// --------- pass1: e = leaky_relu(el[src]+er[dst]); m[dst] = max --------------
// (Reference's isfinite/s==0 fixups only matter for nodes with no in-edges,
//  which no edge ever reads -> dead code here.)

__global__ void edge_pass1(const float* __restrict__ el,
                           const float* __restrict__ er,
                           const int* __restrict__ src,
                           const int* __restrict__ dst,
                           float* __restrict__ ebuf, float* __restrict__ m,
                           int E, int H) {
    int e = blockIdx.x * blockDim.x + threadIdx.x;
    if (e >= E) return;
    int s = src[e], d = dst[e];
    #pragma unroll 4
    for (int h = 0; h < H; ++h) {
        float v = el[s * H + h] + er[d * H + h];
        v = v > 0.f ? v : v * NEG_SLOPE;
        ebuf[(size_t)e * H + h] = v;
        atomicMaxF(&m[d * H + h], v);
    }
}

// --------- pass2: ex = exp(e - m[dst]); s[dst] += ex (ex stored in place) ----

__global__ void edge_pass2(float* __restrict__ ebuf,
                           const float* __restrict__ m,
                           const int* __restrict__ dst,
                           float* __restrict__ ssum, int E, int H) {
    int e = blockIdx.x * blockDim.x + threadIdx.x;
    if (e >= E) return;
    int d = dst[e];
    #pragma unroll 4
    for (int h = 0; h < H; ++h) {
        float ex = __expf(ebuf[(size_t)e * H + h] - m[d * H + h]);
        ebuf[(size_t)e * H + h] = ex;
        atomicAdd(&ssum[d * H + h], ex);
    }
}

// --------- pass3: h_next[dst] += (ex/s[dst]) * feat[src] ---------------------
// One wave32 per edge; lane covers a float4 of the 128-wide row (coalesced
// 512B gather), scatter via global_atomic_add_f32 into an L2-resident buffer.

__global__ __launch_bounds__(256) void edge_pass3(
    const float* __restrict__ ebuf, const float* __restrict__ ssum,
    const float* __restrict__ feat,
    const int* __restrict__ src, const int* __restrict__ dst,
    float* __restrict__ hn, int E, int H, int D) {
    int lane = threadIdx.x & 31;
    int wv   = threadIdx.x >> 5;
    int e    = blockIdx.x * 8 + wv;
    if (e >= E) return;
    int s = src[e], d = dst[e];
    int d0 = lane * 4;
    int head = d0 / D;                       // H=4,D=32 or H=1,D=128
    float a = ebuf[(size_t)e * H + head] / ssum[d * H + head];
    float4 f = *(const float4*)(feat + (size_t)s * 128 + d0);
    float* out = hn + (size_t)d * 128 + d0;
    atomicAdd(out + 0, a * f.x);
    atomicAdd(out + 1, a * f.y);
    atomicAdd(out + 2, a * f.z);
    atomicAdd(out + 3, a * f.w);
}

// --------------------------- ELU (in place) ---------------------------------

__global__ void elu_kernel(float* __restrict__ h, size_t tot) {
    size_t tid = (size_t)blockIdx.x * blockDim.x + threadIdx.x;
    if (tid >= tot) return;
    float v = h[tid];
    h[tid] = v > 0.f ? v : expm1f(v);
}

// --------------------------- host-side layer --------------------------------

static void gat_layer(const float* h_in, const float* W, const float* bias,
                      const float* al, const float* ar,
                      const int* src, const int* dst, float* h_out,
                      float* feat, float* ebuf, float* el, float* er,
                      float* m, float* s, int n, int E, int H, int D,
                      bool do_elu, hipStream_t stream) {
    gemm128_wmma<<<(n + 127) / 128, 256, 0, stream>>>(h_in, W, feat, n);
    scores_kernel<<<(n * H + 255) / 256, 256, 0, stream>>>(feat, al, ar, el, er, n, H, D);
    init_kernel<<<(int)(((size_t)n * 128 + 255) / 256), 256, 0, stream>>>(h_in, bias, h_out, m, s, n, H);
    edge_pass1<<<(E + 255) / 256, 256, 0, stream>>>(el, er, src, dst, ebuf, m, E, H);
    edge_pass2<<<(E + 255) / 256, 256, 0, stream>>>(ebuf, m, dst, s, E, H);
    edge_pass3<<<(E + 7) / 8, 256, 0, stream>>>(ebuf, s, feat, src, dst, h_out, E, H, D);
    if (do_elu)
        elu_kernel<<<(int)(((size_t)n * 128 + 255) / 256), 256, 0, stream>>>(h_out, (size_t)n * 128);
}

// --------------------------- entry point ------------------------------------

extern "C" void kernel_launch(void* const* d_in, const int* in_sizes, int n_in,
                              void* d_out, int out_size, void* d_ws, size_t ws_size,
                              hipStream_t stream) {
    const float* nf   = (const float*)d_in[0];   // [n,128]
    const float* Ws   = (const float*)d_in[1];   // [4,128,128]
    const float* bs   = (const float*)d_in[2];   // [4,128]
    const float* al_h = (const float*)d_in[3];   // [3,4,32]
    const float* ar_h = (const float*)d_in[4];   // [3,4,32]
    const float* al_o = (const float*)d_in[5];   // [1,128]
    const float* ar_o = (const float*)d_in[6];   // [1,128]
    const int*   src  = (const int*)d_in[7];     // [E]
    const int*   dst  = (const int*)d_in[8];     // [E]

    const int n = in_sizes[0] / 128;
    const int E = in_sizes[7];
    float* out = (float*)d_out;

    // Workspace layout (floats): feat | hA | ebuf(E*4) | el | er | m | s
    float* feat = (float*)d_ws;
    float* hA   = feat + (size_t)n * 128;
    float* ebuf = hA + (size_t)n * 128;
    float* el   = ebuf + (size_t)E * 4;
    float* er   = el + (size_t)n * 4;
    float* m    = er + (size_t)n * 4;
    float* s    = m  + (size_t)n * 4;

    // 3 hidden layers (H=4, D=32) + output layer (H=1, D=128).
    // Ping-pong: nf -> hA -> out -> hA -> out (mean over 1 head == identity).
    gat_layer(nf,  Ws + 0 * 16384, bs + 0 * 128, al_h + 0 * 128, ar_h + 0 * 128,
              src, dst, hA,  feat, ebuf, el, er, m, s, n, E, 4, 32, true,  stream);
    gat_layer(hA,  Ws + 1 * 16384, bs + 1 * 128, al_h + 1 * 128, ar_h + 1 * 128,
              src, dst, out, feat, ebuf, el, er, m, s, n, E, 4, 32, true,  stream);
    gat_layer(out, Ws + 2 * 16384, bs + 2 * 128, al_h + 2 * 128, ar_h + 2 * 128,
              src, dst, hA,  feat, ebuf, el, er, m, s, n, E, 4, 32, true,  stream);
    gat_layer(hA,  Ws + 3 * 16384, bs + 3 * 128, al_o,           ar_o,
              src, dst, out, feat, ebuf, el, er, m, s, n, E, 1, 128, false, stream);
}